// Model_11879879542757
// MI455X (gfx1250) — compile-verified
//
#include <hip/hip_runtime.h>

// DenseHashTable lookup == lower_bound on a sorted int32 key array + value gather.
// MI455X strategy:
//  * 64 MB table (keys+values) is fully resident in the 192 MB global L2 ->
//    every probe after warm-up is an L2 hit; problem is L2-latency bound.
//  * Interpolation-guided lower_bound (keys uniform in [1,2^31)) cuts the
//    dependent probe chain from 23 to ~5 per query; invariant-preserving mid
//    selection keeps exact searchsorted(side='left') semantics incl. duplicates.
//  * 8 interleaved queries per thread, BRANCH-FREE loop body: converged
//    queries are a fixed point under clamped re-probing, so no per-query
//    exec masking -> the 8 probe loads cluster into one load clause per
//    iteration (max MLP), updates are v_cndmask selects.
//  * Interpolated mid uses hardware v_rcp_f32 (1 ulp): any clamped mid
//    preserves the lower_bound invariant, so precision is free.
//  * Non-temporal hints on the streaming query/output traffic so it does not
//    evict the L2-resident table (CDNA5 TH=NT on global_load/global_store).

typedef __attribute__((ext_vector_type(4))) int   v4i;
typedef __attribute__((ext_vector_type(4))) float v4f;

#define QPT 8  // queries per thread (interleaved searches for MLP)

__global__ __launch_bounds__(256) void ht_lookup_kernel(
    const int* __restrict__ query,
    const int* __restrict__ keys,
    const int* __restrict__ vals,
    float* __restrict__ out,
    int n, int K)
{
    const int t    = blockIdx.x * blockDim.x + threadIdx.x;
    const int base = t * QPT;
    if (base >= n) return;

    // ---- load 8 queries (streaming, non-temporal, 2x b128) ----
    int q[QPT];
    if (base + QPT - 1 < n) {
        v4i qa = __builtin_nontemporal_load((const v4i*)(query + base));
        v4i qb = __builtin_nontemporal_load((const v4i*)(query + base + 4));
        q[0] = qa.x; q[1] = qa.y; q[2] = qa.z; q[3] = qa.w;
        q[4] = qb.x; q[5] = qb.y; q[6] = qb.z; q[7] = qb.w;
    } else {
        #pragma unroll
        for (int j = 0; j < QPT; ++j)
            q[j] = (base + j < n) ? query[base + j] : 1;
    }

    // ---- lower_bound state per query ----
    // Invariant: all keys in [0,lo) are < q; all keys in [hi,K) are >= q.
    // kLo = known key value < q below lo; kHi = known key value >= q at hi.
    unsigned lo[QPT], hi[QPT];
    int kLo[QPT], kHi[QPT];
    #pragma unroll
    for (int j = 0; j < QPT; ++j) {
        lo[j] = 0u; hi[j] = (unsigned)K;
        kLo[j] = 0;                 // all queries >= 1
        kHi[j] = 2147483647;        // KEY_SPACE sentinel, >= any query
    }

    // ---- interleaved search: 6 interpolation steps, then bisection ----
    // Branch-free body: converged queries (lo==hi) are a fixed point under
    // clamped re-probe, so no per-query predication is needed. Single
    // wave-level exit test per iteration. Terminates in <= 6 + log2(K)
    // iterations; cap at 48 for absolute safety.
    for (int step = 0; step < 48; ++step) {
        unsigned midc[QPT];
        int k[QPT];

        // phase 1: compute all probe addresses (pure VALU)
        #pragma unroll
        for (int j = 0; j < QPT; ++j) {
            const unsigned span = hi[j] - lo[j];          // may be 0 (converged)
            unsigned off;
            if (step < 6) {
                const float num = (float)(unsigned)(q[j]  - kLo[j]);
                const float den = (float)(unsigned)(kHi[j] - kLo[j]);
                off = (unsigned)(num * __builtin_amdgcn_rcpf(den) * (float)span);
                const unsigned offMax = span ? span - 1u : 0u;
                if (off > offMax) off = offMax;
            } else {
                off = span >> 1;
            }
            unsigned mid = lo[j] + off;                   // == K only when lo==hi==K
            midc[j] = mid < (unsigned)K ? mid : (unsigned)K - 1u;
        }

        // phase 2: issue all 8 independent probes (clusters into one clause)
        #pragma unroll
        for (int j = 0; j < QPT; ++j)
            k[j] = keys[midc[j]];                         // L2-resident probe

        // phase 3: branch-free state update (v_cndmask selects)
        #pragma unroll
        for (int j = 0; j < QPT; ++j) {
            const bool below = (k[j] < q[j]);
            lo[j]  = below ? midc[j] + 1u : lo[j];
            hi[j]  = below ? hi[j]        : midc[j];
            kLo[j] = below ? k[j]         : kLo[j];
            kHi[j] = below ? kHi[j]       : k[j];
        }

        // single wave-level convergence test
        bool any = false;
        #pragma unroll
        for (int j = 0; j < QPT; ++j) any |= (lo[j] < hi[j]);
        if (!any) break;
    }

    // ---- resolve: lo == hi == first index with key >= q.
    // If hi was ever lowered, kHi == keys[lo]; if never lowered, lo == K (miss).
    float r[QPT];
    #pragma unroll
    for (int j = 0; j < QPT; ++j) {
        const unsigned idx = lo[j];
        const bool hit = (idx < (unsigned)K) && (kHi[j] == q[j]);
        int v = -1;
        if (hit) v = vals[idx];                 // predicated gather (L2 hit)
        r[j] = (float)v;
    }

    // ---- store results (streaming, non-temporal, 2x b128) ----
    if (base + QPT - 1 < n) {
        v4f oa; oa.x = r[0]; oa.y = r[1]; oa.z = r[2]; oa.w = r[3];
        v4f ob; ob.x = r[4]; ob.y = r[5]; ob.z = r[6]; ob.w = r[7];
        __builtin_nontemporal_store(oa, (v4f*)(out + base));
        __builtin_nontemporal_store(ob, (v4f*)(out + base + 4));
    } else {
        #pragma unroll
        for (int j = 0; j < QPT; ++j)
            if (base + j < n) out[base + j] = r[j];
    }
}

extern "C" void kernel_launch(void* const* d_in, const int* in_sizes, int n_in,
                              void* d_out, int out_size, void* d_ws, size_t ws_size,
                              hipStream_t stream) {
    const int* query = (const int*)d_in[0];
    const int* keys  = (const int*)d_in[1];
    const int* vals  = (const int*)d_in[2];
    float*     out   = (float*)d_out;
    const int n = in_sizes[0];   // 4194304 queries
    const int K = in_sizes[1];   // 8388608 table entries

    const int threads  = 256;                       // 8 wave32 waves per block
    const int perBlock = threads * QPT;
    const int blocks   = (n + perBlock - 1) / perBlock;
    hipLaunchKernelGGL(ht_lookup_kernel, dim3(blocks), dim3(threads), 0, stream,
                       query, keys, vals, out, n, K);
}